// SelfAttentionV2_65481071404893
// MI455X (gfx1250) — compile-verified
//
#include <hip/hip_runtime.h>

// 4-qubit statevector sim (lane-parallel trig + 1-thread gate algebra per block)
// followed by 64 MiB broadcast stream-out.
// Bandwidth-bound: 23.3 TB/s -> ~2.8us floor. Store path uses gfx1250
// async LDS->global B128 stores (ASYNCcnt), verified to lower in round 2.

#define QS_BLOCK 256
#define ROWS_PER_THREAD 16

#if defined(__has_builtin)
#if __has_builtin(__builtin_amdgcn_global_store_async_from_lds_b128)
#define HAS_ASYNC_STORE 1
#endif
#endif
#ifndef HAS_ASYNC_STORE
#define HAS_ASYNC_STORE 0
#endif

#if HAS_ASYNC_STORE
// Builtin signature (probe-confirmed round 1->2): (AS1 v4i*, AS3 v4i*, imm offset, imm cpol)
typedef int v4i_gcc __attribute__((vector_size(16)));
typedef __attribute__((address_space(1))) v4i_gcc* gv4i_ptr;
typedef __attribute__((address_space(3))) v4i_gcc* lv4i_ptr;
#endif

__device__ __forceinline__ void apply_rx_pair(float2& a, float2& b, float c, float sn) {
    // [a';b'] = [[c, -i*s],[-i*s, c]] [a;b]
    float2 a0 = a, b0 = b;
    a = make_float2(c * a0.x + sn * b0.y, c * a0.y - sn * b0.x);
    b = make_float2(sn * a0.y + c * b0.x, -sn * a0.x + c * b0.y);
}

// Gate algebra only -- trig values precomputed (tc/ts index k: k=3w+{0,1,2} for
// RX/RY/RZ on wire w, k=12+i for CRX i->i+1). Wire 0 = MSB of state index.
__device__ void simulate_expvals(const float* tc, const float* ts, float ev[4]) {
    float2 s[16];
#pragma unroll
    for (int i = 0; i < 16; ++i) s[i] = make_float2(0.f, 0.f);
    s[0] = make_float2(1.f, 0.f);

#pragma unroll
    for (int w = 0; w < 4; ++w) {
        const int stride = 1 << (3 - w);
        const float cx = tc[w * 3 + 0], sx = ts[w * 3 + 0];
        const float cy = tc[w * 3 + 1], sy = ts[w * 3 + 1];
        const float cz = tc[w * 3 + 2], sz = ts[w * 3 + 2];
        // RX
#pragma unroll
        for (int idx = 0; idx < 16; ++idx) {
            if (idx & stride) continue;
            apply_rx_pair(s[idx], s[idx | stride], cx, sx);
        }
        // RY (real rotation)
#pragma unroll
        for (int idx = 0; idx < 16; ++idx) {
            if (idx & stride) continue;
            float2 a = s[idx], b = s[idx | stride];
            s[idx]          = make_float2(cy * a.x - sy * b.x, cy * a.y - sy * b.y);
            s[idx | stride] = make_float2(sy * a.x + cy * b.x, sy * a.y + cy * b.y);
        }
        // RZ: e^{-i t/2} on bit=0, e^{+i t/2} on bit=1
#pragma unroll
        for (int idx = 0; idx < 16; ++idx) {
            const float ph = (idx & stride) ? sz : -sz;
            float2 a = s[idx];
            s[idx] = make_float2(a.x * cz - a.y * ph, a.x * ph + a.y * cz);
        }
    }
    // CRX chain i -> i+1 (apply RX to target where control bit = 1)
#pragma unroll
    for (int i = 0; i < 3; ++i) {
        const int cb = 1 << (3 - i), tb = 1 << (2 - i);
        const float c = tc[12 + i], sn = ts[12 + i];
#pragma unroll
        for (int idx = 0; idx < 16; ++idx) {
            if (!(idx & cb) || (idx & tb)) continue;
            apply_rx_pair(s[idx], s[idx | tb], c, sn);
        }
    }
    ev[0] = ev[1] = ev[2] = ev[3] = 0.f;
#pragma unroll
    for (int idx = 0; idx < 16; ++idx) {
        const float p = s[idx].x * s[idx].x + s[idx].y * s[idx].y;
#pragma unroll
        for (int w = 0; w < 4; ++w)
            ev[w] += ((idx >> (3 - w)) & 1) ? -p : p;
    }
}

__global__ __launch_bounds__(QS_BLOCK) void qsim_broadcast_store(
    const float* __restrict__ rot, const float* __restrict__ ent,
    float4* __restrict__ out, long long nrows) {
    __shared__ float trig_c[16];
    __shared__ float trig_s[16];
    __shared__ float4 sval;
    __shared__ float4 tile[QS_BLOCK];  // per-thread 16B replica -> conflict-free async LDS reads

    const int tid = threadIdx.x;

    // Lanes 0..14 compute one sin/cos pair each (one trig latency, not 15 serial).
    if (tid < 15) {
        const float a = 0.5f * ((tid < 12) ? rot[tid] : ent[tid - 12]);
        trig_c[tid] = cosf(a);
        trig_s[tid] = sinf(a);
    }
    __syncthreads();

    if (tid == 0) {
        float ev[4];
        simulate_expvals(trig_c, trig_s, ev);
        sval = make_float4(ev[0], ev[1], ev[2], ev[3]);
    }
    __syncthreads();

    tile[tid] = sval;
    __syncthreads();  // DS writes visible before the async engine reads LDS

    const long long base =
        (long long)blockIdx.x * (long long)(QS_BLOCK * ROWS_PER_THREAD) + tid;

#pragma unroll
    for (int j = 0; j < ROWS_PER_THREAD; ++j) {
        const long long r = base + (long long)j * QS_BLOCK;
        if (r < nrows) {
#if HAS_ASYNC_STORE
            __builtin_amdgcn_global_store_async_from_lds_b128(
                (gv4i_ptr)(unsigned long long)(out + r),
                (lv4i_ptr)(unsigned int)(unsigned long long)(&tile[tid]),
                0, 0);
#else
            unsigned long long ga = (unsigned long long)(out + r);
            unsigned int la = (unsigned int)(unsigned long long)(&tile[tid]);
            asm volatile("global_store_async_from_lds_b128 %0, %1, off"
                         :: "v"(ga), "v"(la) : "memory");
#endif
        }
    }
    // Drain ASYNCcnt (S_ENDPGM also implies wait-idle; explicit for clarity)
    asm volatile("s_wait_asynccnt 0x0" ::: "memory");
}

extern "C" void kernel_launch(void* const* d_in, const int* in_sizes, int n_in,
                              void* d_out, int out_size, void* d_ws, size_t ws_size,
                              hipStream_t stream) {
    (void)in_sizes; (void)n_in; (void)d_ws; (void)ws_size;
    // d_in[0] = inputs (unused by the circuit), d_in[1] = rotation_params (4x3 f32),
    // d_in[2] = entangle_params (3 f32). Output: (BATCH, 4) f32.
    const float* rot = (const float*)d_in[1];
    const float* ent = (const float*)d_in[2];
    float4* out = (float4*)d_out;
    const long long nrows = (long long)out_size / 4;

    const long long rows_per_block = (long long)QS_BLOCK * ROWS_PER_THREAD;
    const int grid = (int)((nrows + rows_per_block - 1) / rows_per_block);

    qsim_broadcast_store<<<grid, QS_BLOCK, 0, stream>>>(rot, ent, out, nrows);
}